// LennardJonesForce_7687991460463
// MI455X (gfx1250) — compile-verified
//
#include <hip/hip_runtime.h>

// CDNA5 / gfx1250: wave32. WMMA f32 16x16x4: A,B = 2 VGPRs (float2), C/D = 8 VGPRs (float8).
typedef __attribute__((ext_vector_type(2))) float v2f_t;
typedef __attribute__((ext_vector_type(8))) float v8f_t;

#define LJ_TPB      256
#define LJ_NBLOCKS  2048          // multiple of 4 (WMMA K-chunks)
#define LJ_PSTRIDE  16            // floats per partial row (11 used, rest zeroed)
#define LJ_NCOMP    11            // energy, virial, vt[9]

__global__ __launch_bounds__(LJ_TPB)
void lj_zero_kernel(float* __restrict__ out, int n) {
  int i = blockIdx.x * blockDim.x + threadIdx.x;
  if (i < n) out[i] = 0.0f;
}

__global__ __launch_bounds__(LJ_TPB)
void lj_edge_kernel(const float* __restrict__ pos,
                    const int*   __restrict__ edge_index,   // [2, E]
                    const float* __restrict__ eps,
                    const float* __restrict__ sig,
                    float*       __restrict__ out,          // [0]=E, [1..3N]=forces, [3N+1]=virial, [3N+2..3N+10]=vt
                    float*       __restrict__ partials,     // [NBLOCKS, 16] or null
                    int E, int N)
{
  constexpr float RC   = 0.5f;
  constexpr float RS   = 0.45f;      // SWITCH_RATIO * RC
  constexpr float INVW = 20.0f;      // 1 / (RC - RS)
  float* __restrict__ forces = out + 1;

  float acc[LJ_NCOMP];
  #pragma unroll
  for (int c = 0; c < LJ_NCOMP; ++c) acc[c] = 0.0f;

  const int tid    = blockIdx.x * blockDim.x + threadIdx.x;
  const int stride = gridDim.x * blockDim.x;

  for (int e = tid; e < E; e += stride) {
    // Read-once streams: non-temporal so they don't evict pos/forces from L2.
    const int   i   = __builtin_nontemporal_load(edge_index + e);
    const int   j   = __builtin_nontemporal_load(edge_index + E + e);
    const float epv = __builtin_nontemporal_load(eps + e);
    const float sgv = __builtin_nontemporal_load(sig + e);

    // Gathers: pos is 1.2 MB, L2/WGP$-resident.
    float rx = pos[3*i+0] - pos[3*j+0];
    float ry = pos[3*i+1] - pos[3*j+1];
    float rz = pos[3*i+2] - pos[3*j+2];
    // minimum image, BOX_LEN = 1; rintf = round-half-to-even, matches jnp.round
    rx -= rintf(rx); ry -= rintf(ry); rz -= rintf(rz);

    const float r2    = fmaxf(rx*rx + ry*ry + rz*rz, 1e-24f);
    const float r     = sqrtf(r2);
    const float inv_r = 1.0f / r;

    const float sr   = sgv * inv_r;
    const float sr2  = sr * sr;
    const float sr6  = sr2 * sr2 * sr2;
    const float sr12 = sr6 * sr6;
    const float U    = 4.0f  * epv * (sr12 - sr6);
    const float Fr   = 24.0f * epv * inv_r * (2.0f * sr12 - sr6);

    // switching: clamped x makes S=1 below rs, S=0 above RC, dS=0 at both ends
    float x = (r - RS) * INVW;
    x = fminf(fmaxf(x, 0.0f), 1.0f);
    const float x2  = x * x;
    const float S   = 1.0f - x * x2 * (10.0f + x * (-15.0f + 6.0f * x));
    const float omx = 1.0f - x;
    const float dS  = -600.0f * x2 * omx * omx;   // (-30x^2+60x^3-30x^4)*20

    const float m  = (r < RC) ? 1.0f : 0.0f;
    const float Ue = U * S * m;
    const float Fm = (Fr * S - U * dS) * m;

    const float fx = Fm * rx * inv_r;
    const float fy = Fm * ry * inv_r;
    const float fz = Fm * rz * inv_r;

    // Scatter into L2-resident forces: native f32 atomics, relaxed/agent scope.
    __hip_atomic_fetch_add(&forces[3*i+0],  fx, __ATOMIC_RELAXED, __HIP_MEMORY_SCOPE_AGENT);
    __hip_atomic_fetch_add(&forces[3*i+1],  fy, __ATOMIC_RELAXED, __HIP_MEMORY_SCOPE_AGENT);
    __hip_atomic_fetch_add(&forces[3*i+2],  fz, __ATOMIC_RELAXED, __HIP_MEMORY_SCOPE_AGENT);
    __hip_atomic_fetch_add(&forces[3*j+0], -fx, __ATOMIC_RELAXED, __HIP_MEMORY_SCOPE_AGENT);
    __hip_atomic_fetch_add(&forces[3*j+1], -fy, __ATOMIC_RELAXED, __HIP_MEMORY_SCOPE_AGENT);
    __hip_atomic_fetch_add(&forces[3*j+2], -fz, __ATOMIC_RELAXED, __HIP_MEMORY_SCOPE_AGENT);

    acc[0]  += Ue;
    acc[1]  += Fm * r;            // virial (mask already folded into Fm)
    acc[2]  += fx * rx;  acc[3]  += fx * ry;  acc[4]  += fx * rz;
    acc[5]  += fy * rx;  acc[6]  += fy * ry;  acc[7]  += fy * rz;
    acc[8]  += fz * rx;  acc[9]  += fz * ry;  acc[10] += fz * rz;
  }

  // wave32 tree reduction
  #pragma unroll
  for (int c = 0; c < LJ_NCOMP; ++c) {
    float v = acc[c];
    #pragma unroll
    for (int off = 16; off > 0; off >>= 1)
      v += __shfl_down(v, off, 32);
    acc[c] = v;
  }

  __shared__ float red[LJ_TPB / 32][LJ_PSTRIDE];
  const int lane = threadIdx.x & 31;
  const int wave = threadIdx.x >> 5;
  if (lane == 0) {
    #pragma unroll
    for (int c = 0; c < LJ_NCOMP; ++c) red[wave][c] = acc[c];
  }
  __syncthreads();

  if (threadIdx.x < LJ_PSTRIDE) {
    float s = 0.0f;
    if (threadIdx.x < LJ_NCOMP) {
      #pragma unroll
      for (int w = 0; w < LJ_TPB / 32; ++w) s += red[w][threadIdx.x];
    }
    if (partials) {
      // cols 11..15 written as 0 so the WMMA reduction sees clean columns
      partials[blockIdx.x * LJ_PSTRIDE + threadIdx.x] = s;
    } else if (threadIdx.x < LJ_NCOMP) {
      float* dst;
      const int c = threadIdx.x;
      if (c == 0)      dst = out;                    // total_energy
      else if (c == 1) dst = out + 3 * N + 1;        // virial
      else             dst = out + 3 * N + 2 + (c - 2); // virial_tensor
      __hip_atomic_fetch_add(dst, s, __ATOMIC_RELAXED, __HIP_MEMORY_SCOPE_AGENT);
    }
  }
}

// Ones-matmul reduction: C[m][n] = sum_k B[k][n] with A == 1 everywhere.
// Invariant to the K-row <-> (lane-half, VGPR) mapping of the B layout, and all
// rows of C are identical, so any C VGPR / lane-half gives the column sums.
__global__ __launch_bounds__(32)
void lj_wmma_reduce_kernel(const float* __restrict__ partials,
                           float* __restrict__ out,
                           int nrows, int N)
{
  const int lane = threadIdx.x;      // single wave32, EXEC all-ones through the loop
  const int col  = lane & 15;        // B column = N = lane % 16 (documented striping)
  const int half = lane >> 4;

  v2f_t a; a[0] = 1.0f; a[1] = 1.0f; // A = ones (16x4)
  v8f_t c = {};

  for (int b = 0; b < nrows; b += 4) {
    v2f_t bv;
    bv[0] = partials[(b + 2 * half + 0) * LJ_PSTRIDE + col];
    bv[1] = partials[(b + 2 * half + 1) * LJ_PSTRIDE + col];
    c = __builtin_amdgcn_wmma_f32_16x16x4_f32(
        /*neg_a=*/false, a, /*neg_b=*/false, bv,
        /*c_mod=*/(short)0, c, /*reuse_a=*/false, /*reuse_b=*/false);
  }

  if (lane < LJ_NCOMP) {
    const float s = c[0];
    if (lane == 0)      out[0] = s;                       // total_energy
    else if (lane == 1) out[3 * N + 1] = s;               // virial
    else                out[3 * N + 2 + (lane - 2)] = s;  // virial_tensor
  }
}

extern "C" void kernel_launch(void* const* d_in, const int* in_sizes, int n_in,
                              void* d_out, int out_size, void* d_ws, size_t ws_size,
                              hipStream_t stream) {
  const float* pos        = (const float*)d_in[0];
  const int*   edge_index = (const int*)  d_in[1];
  const float* eps        = (const float*)d_in[2];
  const float* sig        = (const float*)d_in[3];
  float* out = (float*)d_out;

  const int N = in_sizes[0] / 3;   // atoms
  const int E = in_sizes[2];       // edges

  // Zero all outputs every launch (atomic accumulation + graph replay).
  {
    const int g = (out_size + LJ_TPB - 1) / LJ_TPB;
    lj_zero_kernel<<<g, LJ_TPB, 0, stream>>>(out, out_size);
  }

  const size_t need = (size_t)LJ_NBLOCKS * LJ_PSTRIDE * sizeof(float);
  float* partials = (ws_size >= need) ? (float*)d_ws : nullptr;

  lj_edge_kernel<<<LJ_NBLOCKS, LJ_TPB, 0, stream>>>(
      pos, edge_index, eps, sig, out, partials, E, N);

  if (partials) {
    lj_wmma_reduce_kernel<<<1, 32, 0, stream>>>(partials, out, LJ_NBLOCKS, N);
  }
}